// GCN2_45681272161001
// MI455X (gfx1250) — compile-verified
//
#include <hip/hip_runtime.h>
#include <math.h>

// Problem constants from the reference
#define N_NODES 100000
#define E_EDGES 1600000
#define IN_F    128
#define H_F     64
#define OUT_F   40

typedef __attribute__((ext_vector_type(2))) float v2f;
typedef __attribute__((ext_vector_type(8))) float v8f;

// Full-precision CDNA5 matrix op: D(16x16,f32) = A(16x4,f32) x B(4x16,f32) + C
// 8-arg form: (neg_a, A, neg_b, B, c_mod, C, reuse_a, reuse_b)
__device__ __forceinline__ v8f wmma_f32x4(v2f a, v2f b, v8f c) {
  return __builtin_amdgcn_wmma_f32_16x16x4_f32(false, a, false, b, (short)0, c,
                                               false, false);
}

// ---------------------------------------------------------------------------
// Kernel 1: h = relu(x @ w0 + b0), also copies into h0.
// Block = 128 threads (4 waves). Block computes 16 rows x 64 cols.
// Wave w computes the 16x16 tile at cols [16w, 16w+16). K = 128 in steps of 4.
// ---------------------------------------------------------------------------
__global__ void __launch_bounds__(128)
lin0_relu_kernel(const float* __restrict__ x, const float* __restrict__ w0,
                 const float* __restrict__ b0, float* __restrict__ h,
                 float* __restrict__ h0) {
  __shared__ float lw[IN_F * H_F];  // 32 KB: whole w0 in LDS
  const int tid = threadIdx.x;
  for (int i = tid; i < IN_F * H_F; i += 128) lw[i] = w0[i];
  __syncthreads();

  const int lane  = tid & 31;
  const int wave  = tid >> 5;
  const int row16 = blockIdx.x * 16;
  const int m     = lane & 15;          // A-row / B-col lane index
  const int khalf = (lane >> 4) * 2;    // K sub-offset per ISA layout
  const int col   = wave * 16 + m;      // output column

  const float* xrow = x + (size_t)(row16 + m) * IN_F;
  v8f acc = {};
  for (int k = 0; k < IN_F; k += 4) {
    v2f a, b;
    a.x = xrow[k + khalf];
    a.y = xrow[k + khalf + 1];
    b.x = lw[(k + khalf) * H_F + col];
    b.y = lw[(k + khalf + 1) * H_F + col];
    acc = wmma_f32x4(a, b, acc);
  }

  const float bias  = b0[col];
  const int   rbase = row16 + (lane >> 4) * 8;  // C layout: M = g + 8*(lane>=16)
#pragma unroll
  for (int g = 0; g < 8; ++g) {
    float v = acc[g] + bias;
    v = v > 0.f ? v : 0.f;
    size_t idx = (size_t)(rbase + g) * H_F + col;
    h[idx]  = v;
    h0[idx] = v;
  }
}

// ---------------------------------------------------------------------------
// Kernel 2: zero-fill (float4 vectorized)
// ---------------------------------------------------------------------------
__global__ void __launch_bounds__(256)
fill_zero_kernel(float4* __restrict__ p, int n4) {
  int i = blockIdx.x * blockDim.x + threadIdx.x;
  if (i < n4) p[i] = make_float4(0.f, 0.f, 0.f, 0.f);
}

// ---------------------------------------------------------------------------
// Kernel 3: agg[dst] += h[src]  (16 threads/edge, float4 chunk each).
// h and agg are 25.6 MB each -> L2-resident atomics on MI455X (192 MB L2).
// ---------------------------------------------------------------------------
__global__ void __launch_bounds__(256)
scatter_add_kernel(const int* __restrict__ src, const int* __restrict__ dst,
                   const float* __restrict__ h, float* __restrict__ agg,
                   int nE) {
  int tid = blockIdx.x * blockDim.x + threadIdx.x;
  int e = tid >> 4;
  if (e >= nE) return;
  int q = (tid & 15) * 4;
  int s = src[e];
  int d = dst[e];
  const float4 v = *(const float4*)(h + (size_t)s * H_F + q);
  float* ap = agg + (size_t)d * H_F + q;
  atomicAdd(ap + 0, v.x);
  atomicAdd(ap + 1, v.y);
  atomicAdd(ap + 2, v.z);
  atomicAdd(ap + 3, v.w);
}

// ---------------------------------------------------------------------------
// Kernel 4: mix = 0.9*agg + 0.1*h0 (initial residual), float4 vectorized
// ---------------------------------------------------------------------------
__global__ void __launch_bounds__(256)
residual_mix_kernel(const float4* __restrict__ agg,
                    const float4* __restrict__ h0, float4* __restrict__ mix,
                    int n4) {
  int i = blockIdx.x * blockDim.x + threadIdx.x;
  if (i >= n4) return;
  float4 a = agg[i];
  float4 b = h0[i];
  mix[i] = make_float4(0.9f * a.x + 0.1f * b.x, 0.9f * a.y + 0.1f * b.y,
                       0.9f * a.z + 0.1f * b.z, 0.9f * a.w + 0.1f * b.w);
}

// ---------------------------------------------------------------------------
// Kernel 5: h = relu((1-beta)*mix + beta*(mix @ W)), W is 64x64 in LDS.
// Same WMMA tiling as kernel 1, K = 64.
// ---------------------------------------------------------------------------
__global__ void __launch_bounds__(128)
gcn_layer_mm_kernel(const float* __restrict__ mix, const float* __restrict__ W,
                    float* __restrict__ h, float beta) {
  __shared__ float lw[H_F * H_F];  // 16 KB
  const int tid = threadIdx.x;
  for (int i = tid; i < H_F * H_F; i += 128) lw[i] = W[i];
  __syncthreads();

  const int lane  = tid & 31;
  const int wave  = tid >> 5;
  const int row16 = blockIdx.x * 16;
  const int m     = lane & 15;
  const int khalf = (lane >> 4) * 2;
  const int col   = wave * 16 + m;

  const float* orow = mix + (size_t)(row16 + m) * H_F;
  v8f acc = {};
  for (int k = 0; k < H_F; k += 4) {
    v2f a, b;
    a.x = orow[k + khalf];
    a.y = orow[k + khalf + 1];
    b.x = lw[(k + khalf) * H_F + col];
    b.y = lw[(k + khalf + 1) * H_F + col];
    acc = wmma_f32x4(a, b, acc);
  }

  const float cb    = 1.0f - beta;
  const int   rbase = row16 + (lane >> 4) * 8;
#pragma unroll
  for (int g = 0; g < 8; ++g) {
    size_t idx = (size_t)(rbase + g) * H_F + col;
    float o = mix[idx];
    float v = cb * o + beta * acc[g];
    h[idx] = v > 0.f ? v : 0.f;
  }
}

// ---------------------------------------------------------------------------
// Kernel 6: logits = h @ w1 + b1 (w1 zero-padded to 48 cols in LDS),
// then per-row log_softmax over the 40 real outputs.
// Block = 128 threads; waves 0..2 compute the 3 column tiles via WMMA,
// logits parked in LDS, then 16 threads finish log-softmax per row.
// ---------------------------------------------------------------------------
__global__ void __launch_bounds__(128)
final_softmax_kernel(const float* __restrict__ h, const float* __restrict__ w1,
                     const float* __restrict__ b1, float* __restrict__ outp) {
  __shared__ float lw[H_F * 48];     // 12 KB padded weights
  __shared__ float logits[16][48];   // 3 KB
  const int tid = threadIdx.x;
  for (int i = tid; i < H_F * 48; i += 128) {
    int r = i / 48, c = i % 48;
    lw[i] = (c < OUT_F) ? w1[r * OUT_F + c] : 0.f;
  }
  __syncthreads();

  const int lane  = tid & 31;
  const int wave  = tid >> 5;
  const int row16 = blockIdx.x * 16;
  const int m     = lane & 15;
  const int khalf = (lane >> 4) * 2;

  if (wave < 3) {  // wave-uniform branch: EXEC all-1s within each WMMA wave
    const int col = wave * 16 + m;
    const float* hrow = h + (size_t)(row16 + m) * H_F;
    v8f acc = {};
    for (int k = 0; k < H_F; k += 4) {
      v2f a, b;
      a.x = hrow[k + khalf];
      a.y = hrow[k + khalf + 1];
      b.x = lw[(k + khalf) * 48 + col];
      b.y = lw[(k + khalf + 1) * 48 + col];
      acc = wmma_f32x4(a, b, acc);
    }
    const float bias = (col < OUT_F) ? b1[col] : 0.f;
    const int   rb   = (lane >> 4) * 8;
#pragma unroll
    for (int g = 0; g < 8; ++g) logits[rb + g][col] = acc[g] + bias;
  }
  __syncthreads();

  if (tid < 16) {
    int row = row16 + tid;
    float mx = -3.402823466e38f;
    for (int j = 0; j < OUT_F; ++j) mx = fmaxf(mx, logits[tid][j]);
    float s = 0.f;
    for (int j = 0; j < OUT_F; ++j) s += __expf(logits[tid][j] - mx);
    float lse = mx + __logf(s);
    float* op = outp + (size_t)row * OUT_F;
    for (int j = 0; j < OUT_F; ++j) op[j] = logits[tid][j] - lse;
  }
}

// ---------------------------------------------------------------------------
extern "C" void kernel_launch(void* const* d_in, const int* in_sizes, int n_in,
                              void* d_out, int out_size, void* d_ws,
                              size_t ws_size, hipStream_t stream) {
  const float* x  = (const float*)d_in[0];
  const int*   ei = (const int*)d_in[1];  // edge_index [2, E]
  const float* w0 = (const float*)d_in[2];
  const float* b0 = (const float*)d_in[3];
  const float* w1 = (const float*)d_in[4];
  const float* b1 = (const float*)d_in[5];
  const float* cw = (const float*)d_in[6];  // [L, H, H]

  const int* src = ei;
  const int* dst = ei + E_EDGES;

  const size_t NH = (size_t)N_NODES * H_F;  // 6.4M floats = 25.6 MB each
  float* h0  = (float*)d_ws;
  float* h   = h0 + NH;
  float* agg = h + NH;
  float* mix = agg + NH;

  const int nb16 = N_NODES / 16;  // 6250 (exact)
  const int n4   = (int)(NH / 4);

  lin0_relu_kernel<<<nb16, 128, 0, stream>>>(x, w0, b0, h, h0);

  for (int layer = 0; layer < 4; ++layer) {
    fill_zero_kernel<<<(n4 + 255) / 256, 256, 0, stream>>>((float4*)agg, n4);
    scatter_add_kernel<<<(E_EDGES * 16 + 255) / 256, 256, 0, stream>>>(
        src, dst, h, agg, E_EDGES);
    residual_mix_kernel<<<(n4 + 255) / 256, 256, 0, stream>>>(
        (const float4*)agg, (const float4*)h0, (float4*)mix, n4);
    float beta = logf(0.5f / (float)(layer + 1) + 1.0f);
    gcn_layer_mm_kernel<<<nb16, 128, 0, stream>>>(
        mix, cw + (size_t)layer * H_F * H_F, h, beta);
  }

  final_softmax_kernel<<<nb16, 128, 0, stream>>>(h, w1, b1, (float*)d_out);
}